// GCNModel_58789512347897
// MI455X (gfx1250) — compile-verified
//
#include <hip/hip_runtime.h>

#define F_IN 128
#define HID  128
#define EMB  64

typedef float v2f __attribute__((ext_vector_type(2)));
typedef float v8f __attribute__((ext_vector_type(8)));

static inline int cdiv(int a, int b) { return (a + b - 1) / b; }

// ---------------- async global->LDS helpers (gfx1250, raw ISA) ----------------
// Generic LDS pointers are {SHARED_BASE, offset[31:0]}, so the low 32 bits are
// the wave-relative LDS byte address the instruction's VDST operand expects.
__device__ __forceinline__ void async_copy16(const float* gp, float* lp) {
  unsigned lds_addr = (unsigned)(size_t)lp;
  unsigned long long gaddr = (unsigned long long)(size_t)gp;
  asm volatile("global_load_async_to_lds_b128 %0, %1, off"
               :: "v"(lds_addr), "v"(gaddr) : "memory");
}

__device__ __forceinline__ void wait_async0() {
  asm volatile("s_wait_asynccnt 0x0" ::: "memory");
}

// ---------- degree / norm ----------
__global__ void k_fill1(float* p, int n) {
  int i = blockIdx.x * blockDim.x + threadIdx.x;
  if (i < n) p[i] = 1.0f;  // self-loop contributes 1 to every node's degree
}

__global__ void k_deg_edges(const int* __restrict__ dst, float* __restrict__ deg, int ne) {
  int e = blockIdx.x * blockDim.x + threadIdx.x;
  if (e < ne) atomicAdd(&deg[dst[e]], 1.0f);
}

__global__ void k_rsqrt(float* p, int n) {
  int i = blockIdx.x * blockDim.x + threadIdx.x;
  if (i < n) p[i] = rsqrtf(p[i]);  // deg >= 1 always (self loops)
}

// ---------- dense transform: H = X @ W via V_WMMA_F32_16X16X4_F32 ----------
// Persistent blocks: each wave keeps its 16-column B slice in 64 VGPRs and
// iterates over TPB_TILES row tiles; the 16xK A strip is double-buffered in
// LDS and staged with async global->LDS B128 copies (ASYNCcnt).
template<int K, int NCOL, int TPB_TILES>
__global__ __launch_bounds__(256)
void k_gemm_wmma(const float* __restrict__ X, const float* __restrict__ W,
                 float* __restrict__ H, int nrows, int ntiles) {
  constexpr int LDSROW = K + 4;            // 132 floats: 16B aligned, conflict-free
  __shared__ float As[2][16][LDSROW];
  const int tid     = threadIdx.x;
  const int lane    = tid & 31;
  const int colBase = (tid >> 5) * 16;     // wave id -> output column tile
  const int m       = lane & 15;           // A row / B col within tile
  const int kOff    = (lane >> 4) * 2;     // lanes 0-15: K=0,1 ; lanes 16-31: K=2,3

  const int tileBeg = blockIdx.x * TPB_TILES;
  if (tileBeg >= ntiles) return;
  const int tileEnd = (tileBeg + TPB_TILES < ntiles) ? tileBeg + TPB_TILES : ntiles;

  // Hoist this wave's B columns into registers: K/4 v2f fragments (64 VGPRs @K=128)
  v2f bfrag[K / 4];
  #pragma unroll
  for (int kk = 0; kk < K / 4; ++kk) {
    bfrag[kk].x = W[(size_t)(kk * 4 + kOff)     * NCOL + colBase + m];
    bfrag[kk].y = W[(size_t)(kk * 4 + kOff + 1) * NCOL + colBase + m];
  }

  // stage a 16xK strip of X into As[buf] via async copies (16B chunks)
  auto stage = [&](int tile, int buf) {
    constexpr int CHUNKS = 16 * (K / 4);
    for (int ch = tid; ch < CHUNKS; ch += blockDim.x) {
      int r = ch / (K / 4);
      int c = (ch - r * (K / 4)) * 4;
      int gr = tile * 16 + r;
      if (gr >= nrows) gr = nrows - 1;  // only rows < nrows are ever stored
      async_copy16(X + (size_t)gr * K + c, &As[buf][r][c]);
    }
  };

  stage(tileBeg, 0);

  for (int t = tileBeg; t < tileEnd; ++t) {
    const int buf = (t - tileBeg) & 1;
    wait_async0();        // this wave's staged data has landed in LDS
    __syncthreads();      // all waves' data visible; all done reading buf^1
    if (t + 1 < tileEnd) stage(t + 1, buf ^ 1);

    v8f acc = {};
    #pragma unroll
    for (int kk = 0; kk < K / 4; ++kk) {
      v2f a;
      a.x = As[buf][m][kk * 4 + kOff];
      a.y = As[buf][m][kk * 4 + kOff + 1];
      acc = __builtin_amdgcn_wmma_f32_16x16x4_f32(false, a, false, bfrag[kk],
                                                  (short)0, acc, false, false);
    }

    // C/D layout: lane holds col n=lane&15; VGPR r holds row r + (lane>>4)*8
    const int col   = colBase + (lane & 15);
    const int rBase = t * 16 + (lane >> 4) * 8;
    #pragma unroll
    for (int r = 0; r < 8; ++r) {
      int gr = rBase + r;
      if (gr < nrows) H[(size_t)gr * NCOL + col] = acc[r];
    }
    // next iteration's barrier orders the restaging of this buffer
  }
}

// ---------- aggregation ----------
// agg[i] = h[i] * dinv[i]^2 + bias   (self-loop term + post-sum bias)
template<int F>
__global__ void k_agg_init(const float* __restrict__ H, const float* __restrict__ dinv,
                           const float* __restrict__ bias, float* __restrict__ agg, int n) {
  int i = blockIdx.x * blockDim.x + threadIdx.x;
  if (i >= n * F) return;
  int node = i / F, f = i - node * F;
  float di = dinv[node];
  agg[i] = H[i] * di * di + bias[f];
}

// One wave per edge; F/32 features per lane; float atomic adds into agg[dst].
template<int F>
__global__ void k_scatter(const float* __restrict__ H, const int* __restrict__ src,
                          const int* __restrict__ dst, const float* __restrict__ dinv,
                          float* __restrict__ agg, int ne) {
  int g = blockIdx.x * blockDim.x + threadIdx.x;
  int e = g >> 5;
  int lane = g & 31;
  if (e >= ne) return;
  int s = src[e], d = dst[e];
  float nrm = dinv[s] * dinv[d];
  constexpr int VPT = F / 32;
  const float* hrow = H   + (size_t)s * F + lane * VPT;
  float*       arow = agg + (size_t)d * F + lane * VPT;
  #pragma unroll
  for (int i = 0; i < VPT; ++i)
    atomicAdd(&arow[i], hrow[i] * nrm);
}

// ---------- batch norm (training-mode batch stats) + ReLU ----------
__global__ void k_bn_zero(float* sums, float* sumsq) {
  int i = threadIdx.x;
  if (i < HID) { sums[i] = 0.0f; sumsq[i] = 0.0f; }
}

__global__ void k_bn_stats(const float* __restrict__ X, float* __restrict__ sums,
                           float* __restrict__ sumsq, int n) {
  int col  = threadIdx.x;              // 128 threads: one column each
  int row0 = blockIdx.x * 256;
  float s = 0.0f, s2 = 0.0f;
  for (int r = 0; r < 256; ++r) {
    int row = row0 + r;
    if (row < n) {
      float v = X[(size_t)row * HID + col];
      s += v; s2 += v * v;
    }
  }
  atomicAdd(&sums[col], s);
  atomicAdd(&sumsq[col], s2);
}

__global__ void k_bn_relu(float* __restrict__ X, const float* __restrict__ sums,
                          const float* __restrict__ sumsq, const float* __restrict__ g,
                          const float* __restrict__ be, int n) {
  int i = blockIdx.x * blockDim.x + threadIdx.x;
  if (i >= n * HID) return;
  int f = i & (HID - 1);
  float inv_n = 1.0f / (float)n;
  float mu  = sums[f] * inv_n;
  float var = fmaxf(sumsq[f] * inv_n - mu * mu, 0.0f);
  float v = (X[i] - mu) * rsqrtf(var + 1e-5f) * g[f] + be[f];
  X[i] = v > 0.0f ? v : 0.0f;
}

// ---------- orchestration ----------
extern "C" void kernel_launch(void* const* d_in, const int* in_sizes, int n_in,
                              void* d_out, int out_size, void* d_ws, size_t ws_size,
                              hipStream_t stream) {
  const float* x   = (const float*)d_in[0];
  const int*   ei  = (const int*)d_in[1];
  const float* W1  = (const float*)d_in[2];
  const float* b1  = (const float*)d_in[3];
  const float* g1  = (const float*)d_in[4];
  const float* be1 = (const float*)d_in[5];
  const float* W2  = (const float*)d_in[6];
  const float* b2  = (const float*)d_in[7];
  const float* g2  = (const float*)d_in[8];
  const float* be2 = (const float*)d_in[9];
  const float* W3  = (const float*)d_in[10];
  const float* b3  = (const float*)d_in[11];

  const int N = in_sizes[0] / F_IN;   // 100000
  const int E = in_sizes[1] / 2;      // 1600000
  const int* src = ei;
  const int* dst = ei + E;

  // workspace carve-out (256B aligned)
  char* ws = (char*)d_ws;
  size_t off = 0;
  auto carve = [&](size_t bytes) -> void* {
    void* p = ws + off;
    off = (off + bytes + 255) & ~(size_t)255;
    return p;
  };
  float* dinv  = (float*)carve((size_t)N * 4);
  float* sums  = (float*)carve(HID * 4);
  float* sumsq = (float*)carve(HID * 4);
  float* bufA  = (float*)carve((size_t)N * HID * 4);
  float* bufB  = (float*)carve((size_t)N * HID * 4);
  if (off > ws_size) return;  // workspace too small; nothing safe to do

  const int TPB = 256;
  const int ntiles = cdiv(N, 16);            // 6250
  constexpr int TILES_PER_BLOCK = 10;        // 625 persistent GEMM blocks
  const int gemmGrid = cdiv(ntiles, TILES_PER_BLOCK);
  float* out = (float*)d_out;

  // --- normalization coefficients ---
  k_fill1    <<<cdiv(N, TPB), TPB, 0, stream>>>(dinv, N);
  k_deg_edges<<<cdiv(E, TPB), TPB, 0, stream>>>(dst, dinv, E);
  k_rsqrt    <<<cdiv(N, TPB), TPB, 0, stream>>>(dinv, N);

  // --- layer 1: conv -> BN -> ReLU ---
  k_gemm_wmma<F_IN, HID, TILES_PER_BLOCK><<<gemmGrid, 256, 0, stream>>>(x, W1, bufA, N, ntiles);
  k_agg_init<HID><<<cdiv(N * HID, TPB), TPB, 0, stream>>>(bufA, dinv, b1, bufB, N);
  k_scatter<HID><<<cdiv(E * 32, TPB), TPB, 0, stream>>>(bufA, src, dst, dinv, bufB, E);
  k_bn_zero <<<1, 256, 0, stream>>>(sums, sumsq);
  k_bn_stats<<<cdiv(N, 256), HID, 0, stream>>>(bufB, sums, sumsq, N);
  k_bn_relu <<<cdiv(N * HID, TPB), TPB, 0, stream>>>(bufB, sums, sumsq, g1, be1, N);

  // --- layer 2: conv -> BN -> ReLU ---
  k_gemm_wmma<HID, HID, TILES_PER_BLOCK><<<gemmGrid, 256, 0, stream>>>(bufB, W2, bufA, N, ntiles);
  k_agg_init<HID><<<cdiv(N * HID, TPB), TPB, 0, stream>>>(bufA, dinv, b2, bufB, N);
  k_scatter<HID><<<cdiv(E * 32, TPB), TPB, 0, stream>>>(bufA, src, dst, dinv, bufB, E);
  k_bn_zero <<<1, 256, 0, stream>>>(sums, sumsq);
  k_bn_stats<<<cdiv(N, 256), HID, 0, stream>>>(bufB, sums, sumsq, N);
  k_bn_relu <<<cdiv(N * HID, TPB), TPB, 0, stream>>>(bufB, sums, sumsq, g2, be2, N);

  // --- layer 3: conv straight into d_out ---
  k_gemm_wmma<HID, EMB, TILES_PER_BLOCK><<<gemmGrid, 128, 0, stream>>>(bufB, W3, bufA, N, ntiles);
  k_agg_init<EMB><<<cdiv(N * EMB, TPB), TPB, 0, stream>>>(bufA, dinv, b3, out, N);
  k_scatter<EMB><<<cdiv(E * 32, TPB), TPB, 0, stream>>>(bufA, src, dst, dinv, out, E);
}